// Attn_conv_90263032693177
// MI455X (gfx1250) — compile-verified
//
#include <hip/hip_runtime.h>
#include <hip/hip_bf16.h>
#include <math.h>

// Problem constants (from the reference)
#define BB     256   // batch
#define CIN    6
#define LL     512   // sequence length
#define C1     32    // conv1 out channels
#define DM     64    // conv2 out channels == model dim
#define NCLS   10
#define NW     8     // waves per attention workgroup

typedef float v2f __attribute__((ext_vector_type(2)));
typedef float v8f __attribute__((ext_vector_type(8)));

// ---------------------------------------------------------------------------
// Kernel 1: conv1 (6->32, k=3, pad=1) + BN(eval) + ReLU.  h1 layout [B][32][L]
// ---------------------------------------------------------------------------
__global__ void ac_conv1_kernel(const float* __restrict__ x,
                                const float* __restrict__ w,
                                const float* __restrict__ bias,
                                const float* __restrict__ g,
                                const float* __restrict__ beta,
                                const float* __restrict__ mean,
                                const float* __restrict__ var,
                                float* __restrict__ h1) {
  int idx = blockIdx.x * blockDim.x + threadIdx.x;
  if (idx >= BB * C1 * LL) return;
  int l  = idx & (LL - 1);
  int co = (idx >> 9) & (C1 - 1);
  int b  = idx >> 14;
  const float* xb = x + (size_t)b * CIN * LL;
  float acc = bias[co];
#pragma unroll
  for (int ci = 0; ci < CIN; ++ci) {
    const float* xr = xb + ci * LL;
    const float* wr = w + (co * CIN + ci) * 3;
    float x0 = (l > 0)      ? xr[l - 1] : 0.0f;
    float x1 = xr[l];
    float x2 = (l < LL - 1) ? xr[l + 1] : 0.0f;
    acc += wr[0] * x0 + wr[1] * x1 + wr[2] * x2;
  }
  float inv = g[co] * rsqrtf(var[co] + 1e-5f);
  float y   = acc * inv + (beta[co] - mean[co] * inv);
  h1[idx] = fmaxf(y, 0.0f);
}

// ---------------------------------------------------------------------------
// Kernel 2: conv2 (32->64, k=3, pad=1) + BN + ReLU, transposed out: h [B][L][64]
// idx = b*L*64 + l*64 + co  -> contiguous stores
// ---------------------------------------------------------------------------
__global__ void ac_conv2_kernel(const float* __restrict__ h1,
                                const float* __restrict__ w,
                                const float* __restrict__ bias,
                                const float* __restrict__ g,
                                const float* __restrict__ beta,
                                const float* __restrict__ mean,
                                const float* __restrict__ var,
                                float* __restrict__ h) {
  int idx = blockIdx.x * blockDim.x + threadIdx.x;
  if (idx >= BB * DM * LL) return;
  int co = idx & (DM - 1);
  int l  = (idx >> 6) & (LL - 1);
  int b  = idx >> 15;
  const float* hb = h1 + (size_t)b * C1 * LL;
  float acc = bias[co];
#pragma unroll 8
  for (int ci = 0; ci < C1; ++ci) {
    const float* xr = hb + ci * LL;
    const float* wr = w + (co * C1 + ci) * 3;
    float x0 = (l > 0)      ? xr[l - 1] : 0.0f;
    float x1 = xr[l];
    float x2 = (l < LL - 1) ? xr[l + 1] : 0.0f;
    acc += wr[0] * x0 + wr[1] * x1 + wr[2] * x2;
  }
  float inv = g[co] * rsqrtf(var[co] + 1e-5f);
  float y   = acc * inv + (beta[co] - mean[co] * inv);
  h[idx] = fmaxf(y, 0.0f);
}

// ---------------------------------------------------------------------------
// Helper: 16x16 f32 tile C += A(16x64) @ B(64x16) with two-accumulator ILP.
// A rows from `arow` (lane-dependent, stride 1 in k), B rows from `brow`
// (w[n][k] layout, stride 1 in k).  kbase per lane = 4c + 2*(lane>>4).
// ---------------------------------------------------------------------------
__device__ __forceinline__ v8f ac_gemm16_k64(const float* __restrict__ arow,
                                             const float* __restrict__ brow,
                                             int lhalf) {
  v8f acc0 = {}, acc1 = {};
#pragma unroll
  for (int c = 0; c < 16; c += 2) {
    {
      const int kb = 4 * c + 2 * lhalf;
      v2f A; A[0] = arow[kb]; A[1] = arow[kb + 1];
      v2f Bf; Bf[0] = brow[kb]; Bf[1] = brow[kb + 1];
      acc0 = __builtin_amdgcn_wmma_f32_16x16x4_f32(false, A, false, Bf,
                                                   (short)0, acc0, false, false);
    }
    {
      const int kb = 4 * (c + 1) + 2 * lhalf;
      v2f A; A[0] = arow[kb]; A[1] = arow[kb + 1];
      v2f Bf; Bf[0] = brow[kb]; Bf[1] = brow[kb + 1];
      acc1 = __builtin_amdgcn_wmma_f32_16x16x4_f32(false, A, false, Bf,
                                                   (short)0, acc1, false, false);
    }
  }
  return acc0 + acc1;
}

// ---------------------------------------------------------------------------
// Kernel 3: fused QKV projection + flash attention, one workgroup per batch.
//   K,V (512x64 f32 each) staged in LDS; per-wave 16x64 LDS scratch for the
//   Q-projection bounce and P C->A layout conversion.
//   All GEMMs are v_wmma_f32_16x16x4_f32 chains (fp32, no precision loss).
// WMMA fragment mapping (wave32):
//   A 16x4 : a[i] = A[M = lane&15][K = kbase + i],   kbase = 4c + 2*(lane>>4)
//   B 4x16 : b[i] = B[K = kbase + i][N = lane&15]
//   C 16x16: c[r] = C[M = r + 8*(lane>>4)][N = lane&15]
// All loops have compile-time-constant trip counts so EXEC stays all-1s
// (WMMA requirement) and no exec-mask loop structure is generated.
// ---------------------------------------------------------------------------
__global__ void __launch_bounds__(NW * 32, 1)
ac_attn_kernel(const float* __restrict__ h,   // [B][L][64]
               const float* __restrict__ wq, const float* __restrict__ bq,
               const float* __restrict__ wk, const float* __restrict__ bk,
               const float* __restrict__ wv, const float* __restrict__ bv,
               float* __restrict__ out) {     // [B][L][64]
  extern __shared__ float smem[];
  float* Ks = smem;                    // 512*64
  float* Vs = Ks + LL * DM;            // 512*64
  float* Wsb = Vs + LL * DM;           // NW * 16*64 per-wave scratch

  const int b     = blockIdx.x;
  const int tid   = threadIdx.x;
  const int wave  = tid >> 5;
  const int lane  = tid & 31;
  const int lhalf = lane >> 4;         // 0 | 1
  const int l16   = lane & 15;
  const float* hb = h + (size_t)b * LL * DM;
  float* ws = Wsb + wave * (16 * DM);

  // ---- Phase 1: K = h@wk^T+bk and V = h@wv^T+bv into LDS --------------------
  // 2 matrices * 32 row-tiles * 4 col-tiles = 256 tiles; each wave does 32
  // (constant trip count -> uniform control flow).
  for (int i = 0; i < 256 / NW; ++i) {
    const int  t    = wave + i * NW;
    const int  isV  = (t >= 128);
    const int  tt   = t & 127;
    const int  lt   = tt >> 2;         // row tile (16 rows)
    const int  nt   = tt & 3;          // col tile (16 cols)
    const float* w    = isV ? wv : wk;
    const float* bias = isV ? bv : bk;
    float*       dst  = isV ? Vs : Ks;

    const float* arow = hb + (size_t)(lt * 16 + l16) * DM;  // A row for this lane
    const float* brow = w  + (size_t)(nt * 16 + l16) * DM;  // B: w[n][k]
    v8f acc = ac_gemm16_k64(arow, brow, lhalf);

    const float bval = bias[nt * 16 + l16];
#pragma unroll
    for (int r = 0; r < 8; ++r) {
      const int row = lt * 16 + r + 8 * lhalf;
      dst[row * DM + nt * 16 + l16] = acc[r] + bval;
    }
  }
  __syncthreads();

  // ---- Phase 2: per-wave flash attention over 16-row Q tiles ---------------
  for (int qi = 0; qi < (LL / 16) / NW; ++qi) {
    const int qt = wave + qi * NW;

    // Q tile projection (scaled by 1/sqrt(64)=0.125) -> ws[16][64]
#pragma unroll
    for (int nt = 0; nt < 4; ++nt) {
      const float* arow = hb + (size_t)(qt * 16 + l16) * DM;
      const float* brow = wq + (size_t)(nt * 16 + l16) * DM;
      v8f acc = ac_gemm16_k64(arow, brow, lhalf);
      const float bval = bq[nt * 16 + l16];
#pragma unroll
      for (int r = 0; r < 8; ++r)
        ws[(r + 8 * lhalf) * DM + nt * 16 + l16] = (acc[r] + bval) * 0.125f;
    }
    // Pull Q A-fragments into registers (16 chunks of 16x4).
    v2f qreg[16];
#pragma unroll
    for (int c = 0; c < 16; ++c) {
      const int kb = 4 * c + 2 * lhalf;
      qreg[c][0] = ws[l16 * DM + kb];
      qreg[c][1] = ws[l16 * DM + kb + 1];
    }

    v8f o0 = {}, o1 = {}, o2 = {}, o3 = {};
    float rmax[8], rsum[8];
#pragma unroll
    for (int r = 0; r < 8; ++r) { rmax[r] = -3.0e38f; rsum[r] = 0.0f; }

    for (int kt = 0; kt < LL / 16; ++kt) {
      // S(16x16) = Q' (16x64) @ K_tile^T (64x16), two-accumulator chain.
      v8f s0 = {}, s1 = {};
      const float* krow = Ks + (size_t)(kt * 16 + l16) * DM;
#pragma unroll
      for (int c = 0; c < 16; c += 2) {
        {
          const int kb = 4 * c + 2 * lhalf;
          v2f Bf; Bf[0] = krow[kb]; Bf[1] = krow[kb + 1];
          s0 = __builtin_amdgcn_wmma_f32_16x16x4_f32(false, qreg[c], false, Bf,
                                                     (short)0, s0, false, false);
        }
        {
          const int kb = 4 * (c + 1) + 2 * lhalf;
          v2f Bf; Bf[0] = krow[kb]; Bf[1] = krow[kb + 1];
          s1 = __builtin_amdgcn_wmma_f32_16x16x4_f32(false, qreg[c + 1], false, Bf,
                                                     (short)0, s1, false, false);
        }
      }
      v8f s = s0 + s1;

      // Online softmax: rows live across 16 lanes of a half-wave.
      float alpha[8];
#pragma unroll
      for (int r = 0; r < 8; ++r) {
        float m = s[r];
        m = fmaxf(m, __shfl_xor(m, 1, 16));
        m = fmaxf(m, __shfl_xor(m, 2, 16));
        m = fmaxf(m, __shfl_xor(m, 4, 16));
        m = fmaxf(m, __shfl_xor(m, 8, 16));
        const float nm = fmaxf(rmax[r], m);
        alpha[r] = __expf(rmax[r] - nm);
        rmax[r]  = nm;
        const float p = __expf(s[r] - nm);
        s[r] = p;
        float ps = p;
        ps += __shfl_xor(ps, 1, 16);
        ps += __shfl_xor(ps, 2, 16);
        ps += __shfl_xor(ps, 4, 16);
        ps += __shfl_xor(ps, 8, 16);
        rsum[r] = rsum[r] * alpha[r] + ps;
      }
#pragma unroll
      for (int r = 0; r < 8; ++r) {
        o0[r] *= alpha[r]; o1[r] *= alpha[r];
        o2[r] *= alpha[r]; o3[r] *= alpha[r];
      }
      // P: C-layout -> A-layout via per-wave LDS bounce (16x16).
#pragma unroll
      for (int r = 0; r < 8; ++r)
        ws[(r + 8 * lhalf) * 16 + l16] = s[r];
      v2f pa[4];
#pragma unroll
      for (int c = 0; c < 4; ++c) {
        const int kb = 4 * c + 2 * lhalf;
        pa[c][0] = ws[l16 * 16 + kb];
        pa[c][1] = ws[l16 * 16 + kb + 1];
      }
      // O(16x64) += P(16x16) @ V_tile(16x64): 4 independent accumulators.
      const float* vbase = Vs + (size_t)kt * 16 * DM;
#pragma unroll
      for (int c = 0; c < 4; ++c) {
        const int k0 = 4 * c + 2 * lhalf;
        v2f B0; B0[0] = vbase[k0 * DM + 0  + l16]; B0[1] = vbase[(k0 + 1) * DM + 0  + l16];
        o0 = __builtin_amdgcn_wmma_f32_16x16x4_f32(false, pa[c], false, B0, (short)0, o0, false, false);
        v2f B1; B1[0] = vbase[k0 * DM + 16 + l16]; B1[1] = vbase[(k0 + 1) * DM + 16 + l16];
        o1 = __builtin_amdgcn_wmma_f32_16x16x4_f32(false, pa[c], false, B1, (short)0, o1, false, false);
        v2f B2; B2[0] = vbase[k0 * DM + 32 + l16]; B2[1] = vbase[(k0 + 1) * DM + 32 + l16];
        o2 = __builtin_amdgcn_wmma_f32_16x16x4_f32(false, pa[c], false, B2, (short)0, o2, false, false);
        v2f B3; B3[0] = vbase[k0 * DM + 48 + l16]; B3[1] = vbase[(k0 + 1) * DM + 48 + l16];
        o3 = __builtin_amdgcn_wmma_f32_16x16x4_f32(false, pa[c], false, B3, (short)0, o3, false, false);
      }
    }
    // Normalize and store O tile.
    float* ob = out + ((size_t)b * LL + qt * 16) * DM;
#pragma unroll
    for (int r = 0; r < 8; ++r) {
      const float inv = 1.0f / rsum[r];
      const int row = r + 8 * lhalf;
      ob[row * DM + 0  + l16] = o0[r] * inv;
      ob[row * DM + 16 + l16] = o1[r] * inv;
      ob[row * DM + 32 + l16] = o2[r] * inv;
      ob[row * DM + 48 + l16] = o3[r] * inv;
    }
  }
}

// ---------------------------------------------------------------------------
// Kernel 4: mean pool over L + FC (64 -> 10). One block (64 threads) per batch.
// ---------------------------------------------------------------------------
__global__ void ac_pool_fc_kernel(const float* __restrict__ attn_out,
                                  const float* __restrict__ fc_w,
                                  const float* __restrict__ fc_b,
                                  float* __restrict__ logits) {
  __shared__ float pooled[DM];
  const int b = blockIdx.x;
  const int tid = threadIdx.x;   // 0..63
  const float* ob = attn_out + (size_t)b * LL * DM;
  float s0 = 0.0f, s1 = 0.0f, s2 = 0.0f, s3 = 0.0f;
  for (int l = 0; l < LL; l += 4) {
    s0 += ob[(l + 0) * DM + tid];
    s1 += ob[(l + 1) * DM + tid];
    s2 += ob[(l + 2) * DM + tid];
    s3 += ob[(l + 3) * DM + tid];
  }
  pooled[tid] = (s0 + s1 + s2 + s3) * (1.0f / (float)LL);
  __syncthreads();
  if (tid < NCLS) {
    float acc = fc_b[tid];
#pragma unroll
    for (int d = 0; d < DM; ++d) acc += pooled[d] * fc_w[tid * DM + d];
    logits[b * NCLS + tid] = acc;
  }
}

// ---------------------------------------------------------------------------
// Launch
// ---------------------------------------------------------------------------
extern "C" void kernel_launch(void* const* d_in, const int* in_sizes, int n_in,
                              void* d_out, int out_size, void* d_ws, size_t ws_size,
                              hipStream_t stream) {
  const float* x       = (const float*)d_in[0];
  const float* conv1_w = (const float*)d_in[1];
  const float* conv1_b = (const float*)d_in[2];
  const float* bn1_g   = (const float*)d_in[3];
  const float* bn1_b   = (const float*)d_in[4];
  const float* bn1_m   = (const float*)d_in[5];
  const float* bn1_v   = (const float*)d_in[6];
  const float* conv2_w = (const float*)d_in[7];
  const float* conv2_b = (const float*)d_in[8];
  const float* bn2_g   = (const float*)d_in[9];
  const float* bn2_b   = (const float*)d_in[10];
  const float* bn2_m   = (const float*)d_in[11];
  const float* bn2_v   = (const float*)d_in[12];
  const float* wq      = (const float*)d_in[13];
  const float* bq      = (const float*)d_in[14];
  const float* wk      = (const float*)d_in[15];
  const float* bk      = (const float*)d_in[16];
  const float* wv      = (const float*)d_in[17];
  const float* bv      = (const float*)d_in[18];
  const float* fc_w    = (const float*)d_in[19];
  const float* fc_b    = (const float*)d_in[20];
  float* logits = (float*)d_out;

  // Workspace layout: h1 [B][32][L] | h [B][L][64] | attn_out [B][L][64]
  float* h1       = (float*)d_ws;
  float* h        = h1 + (size_t)BB * C1 * LL;
  float* attn_out = h  + (size_t)BB * LL * DM;

  {
    const int total = BB * C1 * LL;
    ac_conv1_kernel<<<(total + 255) / 256, 256, 0, stream>>>(
        x, conv1_w, conv1_b, bn1_g, bn1_b, bn1_m, bn1_v, h1);
  }
  {
    const int total = BB * DM * LL;
    ac_conv2_kernel<<<(total + 255) / 256, 256, 0, stream>>>(
        h1, conv2_w, conv2_b, bn2_g, bn2_b, bn2_m, bn2_v, h);
  }
  {
    const size_t smem = (size_t)(2 * LL * DM + NW * 16 * DM) * sizeof(float); // 288 KB
    (void)hipFuncSetAttribute((const void*)ac_attn_kernel,
                              hipFuncAttributeMaxDynamicSharedMemorySize,
                              (int)smem);
    ac_attn_kernel<<<BB, NW * 32, smem, stream>>>(h, wq, bq, wk, bk, wv, bv, attn_out);
  }
  ac_pool_fc_kernel<<<BB, DM, 0, stream>>>(attn_out, fc_w, fc_b, logits);
}